// DilatedKnnGraph_74345883894094
// MI455X (gfx1250) — compile-verified
//
#include <hip/hip_runtime.h>
#include <hip/hip_bf16.h>

// DilatedKnnGraph for MI455X (gfx1250, wave32).
// S=64 strokes, L=1024 pts/stroke, D=256 dims, K_NN=9, DILATION=2 -> top-18, keep every 2nd edge.
// GEMM via V_WMMA_F32_16X16X4_F32 (exact fp32 matrix pipe), dual accumulators to hide WMMA
// latency, fused register-resident per-row top-18 selection.

typedef __attribute__((ext_vector_type(2))) float v2f;
typedef __attribute__((ext_vector_type(8))) float v8f;

#define DIMS 256
#define LPTS 1024
#define KSEL 18            // K_NN * DILATION neighbors found
#define OUTK 9             // kept after ::2 slicing (KSEL even -> even positions)
#define LDS_STRIDE (DIMS + 4)   // 260 floats: (m*260)%64 = m*4 -> 16 lanes hit 16 distinct banks
#define CSTRIDE 36         // 16x32 tile bounce stride: 144 B/row -> 16B aligned for b128 loads

// ---------------- kernel 0: squared norms ----------------
__global__ __launch_bounds__(256)
void sqnorm_kernel(const float* __restrict__ x, float* __restrict__ sq, int npts) {
    int tid = threadIdx.x;
    int p   = blockIdx.x * 32 + (tid >> 3);   // 8 lanes per point
    int l8  = tid & 7;
    if (p >= npts) return;
    const float* row = x + (size_t)p * DIMS;
    float acc = 0.0f;
    #pragma unroll
    for (int k = l8 * 4; k < DIMS; k += 32) { // 8 lanes x float4 = 32 floats / iter
        float4 v = *(const float4*)(row + k);
        acc += v.x * v.x + v.y * v.y + v.z * v.z + v.w * v.w;
    }
    acc += __shfl_down(acc, 4, 8);
    acc += __shfl_down(acc, 2, 8);
    acc += __shfl_down(acc, 1, 8);
    if (l8 == 0) sq[p] = acc;
}

// ---------------- kernel 1: fused WMMA gram + top-18 ----------------
// one wave per (stroke, 16-row tile); grid = S * (L/16) = 4096 blocks of 32 threads
__global__ __launch_bounds__(32)
void knn_wmma_kernel(const float* __restrict__ x, const float* __restrict__ sq,
                     int* __restrict__ out_nn, int* __restrict__ out_ctr) {
    __shared__ float lds_a[16 * LDS_STRIDE];   // 16 row-points  x 256 feats (padded)
    __shared__ float lds_b[32 * LDS_STRIDE];   // 2 x 16 col-points x 256 feats (padded)
    __shared__ float c_lds[16 * CSTRIDE];      // 16x32 neg-dist bounce buffer (b128-aligned)

    const int lane = threadIdx.x;
    const int bid  = blockIdx.x;
    const int s    = bid >> 6;                 // stroke
    const int row0 = (bid & 63) * 16;          // local row base within stroke

    const float* xs  = x  + (size_t)s * LPTS * DIMS;
    const float* sqs = sq + s * LPTS;

    const int hi  = (lane >= 16) ? 1 : 0;      // half-wave select for the K dimension
    const int l16 = lane & 15;

    // stage A rows (16 x 256 f32 = 16 KB) into LDS, coalesced float4
    for (int i = lane * 4; i < 16 * DIMS; i += 32 * 4) {
        int p = i >> 8, k = i & (DIMS - 1);
        float4 v = *(const float4*)(xs + (size_t)(row0 + p) * DIMS + k);
        *(float4*)(&lds_a[p * LDS_STRIDE + k]) = v;
    }

    // per-lane ||row||^2, indexed by accumulator VGPR v: m = v + 8*hi
    float sqm[8];
    #pragma unroll
    for (int v = 0; v < 8; ++v) sqm[v] = sqs[row0 + v + hi * 8];

    // top-18 (sorted descending neg-distance) kept in VGPRs of lanes 0..15
    float topv[KSEL];
    int   topi[KSEL];
    #pragma unroll
    for (int t = 0; t < KSEL; ++t) { topv[t] = -3.4e38f; topi[t] = 0; }

    for (int ct = 0; ct < LPTS / 32; ++ct) {   // two 16-col tiles per iteration
        const int col0 = ct * 32;

        // stage 2 B column tiles (32 x 256 f32 = 32 KB) into LDS, coalesced float4
        for (int i = lane * 4; i < 32 * DIMS; i += 32 * 4) {
            int p = i >> 8, k = i & (DIMS - 1);
            float4 v = *(const float4*)(xs + (size_t)(col0 + p) * DIMS + k);
            *(float4*)(&lds_b[p * LDS_STRIDE + k]) = v;
        }
        // prefetch next pair of tiles (global_prefetch_b8)
        if (ct + 1 < LPTS / 32)
            __builtin_prefetch(xs + (size_t)(col0 + 32) * DIMS + lane * 32, 0, 1);

        // two independent 16x16 gram tiles over K=256: dual accumulators keep
        // two WMMA dependency chains in flight; A fragment shared by both.
        v8f c0 = {}, c1 = {};
        #pragma unroll 8
        for (int kk = 0; kk < DIMS; kk += 4) {
            v2f a  = *(const v2f*)(&lds_a[l16 * LDS_STRIDE + kk + hi * 2]);
            v2f b0 = *(const v2f*)(&lds_b[l16 * LDS_STRIDE + kk + hi * 2]);
            v2f b1 = *(const v2f*)(&lds_b[(16 + l16) * LDS_STRIDE + kk + hi * 2]);
            c0 = __builtin_amdgcn_wmma_f32_16x16x4_f32(
                     false, a, false, b0, (short)0, c0, false, false);
            c1 = __builtin_amdgcn_wmma_f32_16x16x4_f32(
                     false, a, false, b1, (short)0, c1, false, false);
        }

        // neg_adj = 2*inner - ||a||^2 - ||b||^2 ; scatter both tiles so each lane owns a row
        const float sqn0 = sqs[col0 + l16];
        const float sqn1 = sqs[col0 + 16 + l16];
        #pragma unroll
        for (int v = 0; v < 8; ++v) {
            int m = v + hi * 8;                 // C/D layout: VGPR v, lanes>=16 -> M=v+8
            c_lds[m * CSTRIDE + l16]      = 2.0f * c0[v] - sqm[v] - sqn0;
            c_lds[m * CSTRIDE + 16 + l16] = 2.0f * c1[v] - sqm[v] - sqn1;
        }

        // lanes 0..15: bulk-load this row's 32 candidates (8 x ds_load_b128),
        // then register-resident threshold-guarded insertion into sorted top-18.
        if (lane < 16) {
            float4 blk[8];
            #pragma unroll
            for (int t = 0; t < 8; ++t)
                blk[t] = *(const float4*)(&c_lds[lane * CSTRIDE + t * 4]);

            #pragma unroll
            for (int j = 0; j < 32; ++j) {
                float val = ((const float*)blk)[j];   // constant index after unroll
                if (val > topv[KSEL - 1]) {           // strict > keeps lower index on ties
                    topv[KSEL - 1] = val;
                    topi[KSEL - 1] = col0 + j;
                    #pragma unroll
                    for (int t = KSEL - 1; t > 0; --t) {
                        if (topv[t] > topv[t - 1]) {
                            float tv = topv[t]; topv[t] = topv[t - 1]; topv[t - 1] = tv;
                            int   ti = topi[t]; topi[t] = topi[t - 1]; topi[t - 1] = ti;
                        }
                    }
                }
            }
        }
        // wave re-converges here; EXEC all-ones again before the next WMMA burst
    }

    // emit dilated edges: flat edge e=(p*18+kk); kept iff e even -> positions p*9 + kk/2
    if (lane < 16) {
        const int p = s * LPTS + row0 + lane;          // global center index
        const size_t base = (size_t)p * OUTK;
        #pragma unroll
        for (int t = 0; t < OUTK; ++t) {
            out_nn [base + t] = s * LPTS + topi[2 * t];
            out_ctr[base + t] = p;
        }
    }
}

extern "C" void kernel_launch(void* const* d_in, const int* in_sizes, int n_in,
                              void* d_out, int out_size, void* d_ws, size_t ws_size,
                              hipStream_t stream) {
    const float* x = (const float*)d_in[0];
    const int S    = in_sizes[2];          // 64 strokes (sketch_stroke_num length)
    const int npts = S * LPTS;             // 65536 points

    float* sq = (float*)d_ws;              // npts floats of scratch (256 KB)

    sqnorm_kernel<<<(npts + 31) / 32, 256, 0, stream>>>(x, sq, npts);

    int* out_nn  = (int*)d_out;            // edge_index row 0 (neighbor indices)
    int* out_ctr = out_nn + out_size / 2;  // edge_index row 1 (center indices)
    knn_wmma_kernel<<<S * (LPTS / 16), 32, 0, stream>>>(x, sq, out_nn, out_ctr);
}